// JointEncoding_30786325578173
// MI455X (gfx1250) — compile-verified
//
#include <hip/hip_runtime.h>
#include <hip/hip_bf16.h>
#include <math.h>

typedef float v4f __attribute__((ext_vector_type(4)));

// ---------------------------------------------------------------------------
// Phase 1: transpose one plane [C=32, HW] -> [HW, C=32] via padded LDS tile.
// Source reads are one-shot -> non-temporal (don't cache 102 MB of [C,H,W]
// data we never touch again). Destination writes stay RT so the transposed
// tables become L2-resident for the gather phase.
// ---------------------------------------------------------------------------
__global__ void plane_transpose_kernel(const float* __restrict__ in,
                                       float* __restrict__ out, int HW) {
    __shared__ float tile[64 * 33];  // 64 pixels x 32 channels, padded stride
    const int pix0 = blockIdx.x * 64;

    for (int idx = threadIdx.x; idx < 32 * 64; idx += 256) {
        const int c = idx >> 6;        // 0..31
        const int p = idx & 63;        // contiguous across lanes -> coalesced read
        const int pix = pix0 + p;
        float v = 0.0f;
        if (pix < HW) v = __builtin_nontemporal_load(in + (c * HW + pix));
        tile[p * 33 + c] = v;
    }
    __syncthreads();
    for (int idx = threadIdx.x; idx < 64 * 32; idx += 256) {
        const int p = idx >> 5;
        const int c = idx & 31;        // contiguous across lanes -> coalesced write
        const int pix = pix0 + p;
        if (pix < HW) out[pix * 32 + c] = tile[p * 33 + c];  // RT: keep in L2
    }
}

// ---------------------------------------------------------------------------
// Bilinear sample of one float4 channel-chunk q (0..7) from a [H,W,32] plane
// viewed as v4f[HW*8]. grid_sample semantics: align_corners=True, border pad.
// All indices are 32-bit (max table = 8.4M elements).
// ---------------------------------------------------------------------------
__device__ __forceinline__ v4f bilerp4(const v4f* __restrict__ plane,
                                       int R, float gx, float gy, int q) {
    const float rm1 = (float)(R - 1);
    float fx = (gx + 1.0f) * 0.5f * rm1;
    float fy = (gy + 1.0f) * 0.5f * rm1;
    fx = fminf(fmaxf(fx, 0.0f), rm1);
    fy = fminf(fmaxf(fy, 0.0f), rm1);
    const int x0 = (int)fx;            // floor (fx >= 0)
    const int y0 = (int)fy;
    const float wx = fx - (float)x0;
    const float wy = fy - (float)y0;
    const int x1 = min(x0 + 1, R - 1);
    const int y1 = min(y0 + 1, R - 1);
    const int row0 = y0 * R, row1 = y1 * R;

    // RT loads: these hit the L2-resident tables.
    const v4f f00 = plane[(row0 + x0) * 8 + q];
    const v4f f10 = plane[(row0 + x1) * 8 + q];
    const v4f f01 = plane[(row1 + x0) * 8 + q];
    const v4f f11 = plane[(row1 + x1) * 8 + q];

    const float w00 = (1.0f - wx) * (1.0f - wy);
    const float w10 = wx * (1.0f - wy);
    const float w01 = (1.0f - wx) * wy;
    const float w11 = wx * wy;

    return f00 * w00 + f10 * w10 + f01 * w01 + f11 * w11;
}

// ---------------------------------------------------------------------------
// Phase 2: gather. wave32: 8 lanes per point (each lane owns a float4 channel
// chunk), 4 points per wave. Every corner load is a coalesced b128; output
// stores and coordinate loads are non-temporal so the 512 MB output stream
// does not evict the 102 MB of tables from the 192 MB L2.
// ---------------------------------------------------------------------------
__global__ void triplane_gather_kernel(const float* __restrict__ p_nor,
                                       const v4f* __restrict__ xy0,
                                       const v4f* __restrict__ xz0,
                                       const v4f* __restrict__ yz0,
                                       const v4f* __restrict__ xy1,
                                       const v4f* __restrict__ xz1,
                                       const v4f* __restrict__ yz1,
                                       float* __restrict__ out,
                                       int N, int R0, int R1) {
    const int lane = threadIdx.x & 31;
    const int q    = lane & 7;     // float4 chunk within the 32 channels
    const int sub  = lane >> 3;    // which of 4 points this lane group serves
    const int wavesPerBlock = blockDim.x >> 5;
    const int wave  = blockIdx.x * wavesPerBlock + (threadIdx.x >> 5);
    const int nWave = gridDim.x * wavesPerBlock;
    const int groups = (N + 3) >> 2;

    v4f* __restrict__ o4 = (v4f*)out;

    for (int g = wave; g < groups; g += nWave) {
        // Prefetch next stride's coordinates (global_prefetch_b8 on gfx1250).
        if (g + nWave < groups)
            __builtin_prefetch(p_nor + (g + nWave) * 12, 0, 1);

        const int p = g * 4 + sub;
        if (p < N) {
            const float* gp = p_nor + p * 3;
            const float px = __builtin_nontemporal_load(gp + 0);
            const float py = __builtin_nontemporal_load(gp + 1);
            const float pz = __builtin_nontemporal_load(gp + 2);

            v4f a0 = bilerp4(xy0, R0, px, py, q);
            a0 += bilerp4(xz0, R0, px, pz, q);
            a0 += bilerp4(yz0, R0, py, pz, q);

            v4f a1 = bilerp4(xy1, R1, px, py, q);
            a1 += bilerp4(xz1, R1, px, pz, q);
            a1 += bilerp4(yz1, R1, py, pz, q);

            const int ob = p * 16;             // 64 floats = 16 v4f per point
            __builtin_nontemporal_store(a0, o4 + ob + q);      // level-0 ch 0..31
            __builtin_nontemporal_store(a1, o4 + ob + 8 + q);  // level-1 ch 0..31
        }
    }
}

// ---------------------------------------------------------------------------
// Fallback (only if d_ws is too small for transposed tables): gather directly
// from the [C,H,W] layout. blockDim = 64: one block per point.
// ---------------------------------------------------------------------------
__device__ __forceinline__ float bilerp1(const float* __restrict__ plane,
                                         int R, int c, float gx, float gy) {
    const float rm1 = (float)(R - 1);
    float fx = (gx + 1.0f) * 0.5f * rm1;
    float fy = (gy + 1.0f) * 0.5f * rm1;
    fx = fminf(fmaxf(fx, 0.0f), rm1);
    fy = fminf(fmaxf(fy, 0.0f), rm1);
    const int x0 = (int)fx;
    const int y0 = (int)fy;
    const float wx = fx - (float)x0;
    const float wy = fy - (float)y0;
    const int x1 = min(x0 + 1, R - 1);
    const int y1 = min(y0 + 1, R - 1);
    const float* pc = plane + c * (R * R);
    const float f00 = pc[y0 * R + x0];
    const float f10 = pc[y0 * R + x1];
    const float f01 = pc[y1 * R + x0];
    const float f11 = pc[y1 * R + x1];
    return fmaf(f11, wx * wy,
           fmaf(f01, (1.0f - wx) * wy,
           fmaf(f10, wx * (1.0f - wy), f00 * (1.0f - wx) * (1.0f - wy))));
}

__global__ void triplane_gather_direct_kernel(const float* __restrict__ p_nor,
                                              const float* __restrict__ xy0,
                                              const float* __restrict__ xz0,
                                              const float* __restrict__ yz0,
                                              const float* __restrict__ xy1,
                                              const float* __restrict__ xz1,
                                              const float* __restrict__ yz1,
                                              float* __restrict__ out,
                                              int N, int R0, int R1) {
    const int p = blockIdx.x;
    if (p >= N) return;
    const int j = threadIdx.x;     // 0..63
    const int c = j & 31;
    const float px = p_nor[p * 3 + 0];
    const float py = p_nor[p * 3 + 1];
    const float pz = p_nor[p * 3 + 2];
    float acc;
    if (j < 32) {
        acc  = bilerp1(xy0, R0, c, px, py);
        acc += bilerp1(xz0, R0, c, px, pz);
        acc += bilerp1(yz0, R0, c, py, pz);
    } else {
        acc  = bilerp1(xy1, R1, c, px, py);
        acc += bilerp1(xz1, R1, c, px, pz);
        acc += bilerp1(yz1, R1, c, py, pz);
    }
    __builtin_nontemporal_store(acc, out + ((size_t)p * 64 + j));
}

// ---------------------------------------------------------------------------
// Launch
// ---------------------------------------------------------------------------
extern "C" void kernel_launch(void* const* d_in, const int* in_sizes, int n_in,
                              void* d_out, int out_size, void* d_ws, size_t ws_size,
                              hipStream_t stream) {
    (void)n_in; (void)out_size;
    const float* p_nor = (const float*)d_in[0];
    const float* pl[6];
    size_t elems[6];
    size_t need = 0;
    for (int i = 0; i < 6; ++i) {
        pl[i]    = (const float*)d_in[1 + i];
        elems[i] = (size_t)in_sizes[1 + i];
        need    += elems[i] * sizeof(float);
    }
    const int N = in_sizes[0] / 3;

    const int HW0 = (int)(elems[0] / 32);       // level-0 plane pixels
    const int HW1 = (int)(elems[3] / 32);       // level-1 plane pixels
    const int R0  = (int)lround(sqrt((double)HW0));
    const int R1  = (int)lround(sqrt((double)HW1));

    float* out = (float*)d_out;

    if (ws_size >= need) {
        // Phase 1: transpose all 6 planes into workspace as [H,W,C].
        float* t[6];
        size_t off = 0;
        for (int i = 0; i < 6; ++i) { t[i] = (float*)d_ws + off; off += elems[i]; }
        for (int i = 0; i < 6; ++i) {
            const int HW = (int)(elems[i] / 32);
            const int blocks = (HW + 63) / 64;
            plane_transpose_kernel<<<blocks, 256, 0, stream>>>(pl[i], t[i], HW);
        }
        // Phase 2: coalesced gather (8 lanes/point, 4 points/wave).
        const int groups = (N + 3) >> 2;
        const int wavesPerBlock = 8;            // 256 threads
        int blocks = (groups + wavesPerBlock - 1) / wavesPerBlock;
        if (blocks < 1) blocks = 1;
        triplane_gather_kernel<<<blocks, 256, 0, stream>>>(
            p_nor,
            (const v4f*)t[0], (const v4f*)t[1], (const v4f*)t[2],
            (const v4f*)t[3], (const v4f*)t[4], (const v4f*)t[5],
            out, N, R0, R1);
    } else {
        // Workspace too small: correct strided fallback.
        triplane_gather_direct_kernel<<<N > 0 ? N : 1, 64, 0, stream>>>(
            p_nor, pl[0], pl[1], pl[2], pl[3], pl[4], pl[5], out, N, R0, R1);
    }
}